// Pooling_23038204576405
// MI455X (gfx1250) — compile-verified
//
#include <hip/hip_runtime.h>
#include <math.h>

typedef __attribute__((ext_vector_type(2))) float        v2f;
typedef __attribute__((ext_vector_type(8))) float        v8f;
typedef __attribute__((ext_vector_type(4))) unsigned int v4u;
typedef __attribute__((ext_vector_type(8))) int          v8i;
typedef __attribute__((ext_vector_type(4))) int          v4i;

#define WAVES_PER_BLOCK 8
#define TILES_PER_WAVE  4
#define TILES_PER_BLOCK (WAVES_PER_BLOCK * TILES_PER_WAVE)

// ---------------------------------------------------------------------------
// Kernel 0: zero workspace (accumulator + stats), vectorized float4 stores
// ---------------------------------------------------------------------------
__global__ void zero4_kernel(float4* __restrict__ p, size_t n4) {
    size_t i = (size_t)blockIdx.x * blockDim.x + threadIdx.x;
    if (i < n4) p[i] = make_float4(0.f, 0.f, 0.f, 0.f);
}

// ---------------------------------------------------------------------------
// Kernel 1: sparse conv = TDM-stage W[k] -> gather -> f32 WMMA -> atomic scatter
//   grid = (ceil(tilesPerK / 32), K); blockIdx.y selects the conv offset k.
//   One wave32 handles 4 consecutive 16-row tiles of the kernel map.
//   A frag (16x4 f32):  lane l holds row M=l&15; lanes<16 carry K={0,1},
//                       lanes>=16 carry K={2,3}  (VGPR0=K0/K2, VGPR1=K1/K3)
//   B frag (4x16 f32):  lane l holds col N=l&15; VGPR0=K0|K2, VGPR1=K1|K3
//   C/D  (16x16 f32):   lane l holds N=l&15; VGPR v holds M=v+8*(l>=16)
// ---------------------------------------------------------------------------
__global__ void __launch_bounds__(256)
sparse_conv_wmma_kernel(const float* __restrict__ x,
                        const float* __restrict__ W,
                        const int*   __restrict__ in_idx,
                        const int*   __restrict__ out_idx,
                        float*       __restrict__ acc,
                        int M, int tilesPerK)
{
    __shared__ float Wlds[64 * 64];          // W[k], 16 KB, shared by 8 waves

    const int k    = blockIdx.y;
    const int lane = threadIdx.x & 31;
    const int wid  = threadIdx.x >> 5;
    const float* __restrict__ Wk = W + (size_t)k * 4096u;

    // ---- Stage W[k] into LDS --------------------------------------------
#if __has_builtin(__builtin_amdgcn_tensor_load_to_lds)
    if (wid == 0) {                          // wave-uniform: wave 0 drives TDM
        const unsigned long long ga = (unsigned long long)(uintptr_t)Wk;
        const unsigned int ldsOff   = (unsigned int)(uintptr_t)(void*)Wlds;
        // D# group0: count=1 | lds_addr | global_addr[56:0] | type=2
        v4u g0 = { 1u,
                   ldsOff,
                   (unsigned int)(ga & 0xFFFFFFFFull),
                   (unsigned int)((ga >> 32) & 0x01FFFFFFull) | (2u << 30) };
        // D# group1: data_size=4B; tensor_dim0=4096; tensor_dim1=1;
        //            tile_dim0=4096; tile_dim1=1; dim0_stride=4096
        v8i g1 = { (int)0x00020000,          // data_size=2 (4B) at [17:16]
                   (int)0x10000000,          // tensor_dim0[15:0]=4096 at [63:48]
                   (int)0x00010000,          // tensor_dim1[15:0]=1   at [95:80]
                   (int)0x10000000,          // tile_dim0=4096        at [127:112]
                   1,                        // tile_dim1=1           at [143:128]
                   4096,                     // tensor_dim0_stride    at [191:160]
                   0, 0 };
        v4i gz4  = { 0, 0, 0, 0 };
        v8i gz8  = { 0, 0, 0, 0, 0, 0, 0, 0 };
        // amdgpu-toolchain (clang-23 / therock-10.0) 6-arg form:
        //   (v4u g0, v8i g1, v4i g2, v4i g3, v8i extra, i32 cpol)
        __builtin_amdgcn_tensor_load_to_lds(g0, g1, gz4, gz4, gz8, 0);
        __builtin_amdgcn_s_wait_tensorcnt(0);
    }
#else
    for (int i = threadIdx.x; i < 1024; i += blockDim.x)
        ((float4*)Wlds)[i] = ((const float4*)Wk)[i];
#endif
    __syncthreads();

    // ---- Per-wave tile loop ---------------------------------------------
    const int r  = lane & 15;                // A-row / B-col / D-col in tile
    const int hi = lane >> 4;                // half-wave selector
    const int mtBase = blockIdx.x * TILES_PER_BLOCK + wid * TILES_PER_WAVE;

    for (int j = 0; j < TILES_PER_WAVE; ++j) {
        const int mt = mtBase + j;
        if (mt >= tilesPerK) break;          // wave-uniform exit
        const int m0 = mt << 4;

        int mrow = m0 + r;
        if (mrow >= M) mrow = M - 1;         // clamp: keep EXEC all-ones
        const size_t pairBase = (size_t)k * (size_t)M + (size_t)mrow;
        const int gidx = in_idx[pairBase];
        const int oidx = out_idx[pairBase];
        __builtin_prefetch(&in_idx[pairBase + 16], 0, 0);   // next tile's map

        const float* __restrict__ xrow = x + (size_t)gidx * 64u;

        // Gather this lane's slice of its A row: 16 contiguous float2 loads
        v2f afrag[16];
#pragma unroll
        for (int ks = 0; ks < 16; ++ks)
            afrag[ks] = *(const v2f*)(xrow + ks * 4 + 2 * hi);

#pragma unroll
        for (int nt = 0; nt < 4; ++nt) {     // 16-wide column tiles of C_out
            v8f acc8 = {0.f, 0.f, 0.f, 0.f, 0.f, 0.f, 0.f, 0.f};
#pragma unroll
            for (int ks = 0; ks < 16; ++ks) {           // K-dim 64, steps of 4
                const int c0 = ks * 4 + 2 * hi;         // this lane's B rows
                v2f b;
                b.x = Wlds[c0 * 64 + nt * 16 + r];
                b.y = Wlds[(c0 + 1) * 64 + nt * 16 + r];
                acc8 = __builtin_amdgcn_wmma_f32_16x16x4_f32(
                           /*neg_a=*/false, afrag[ks],
                           /*neg_b=*/false, b,
                           /*c_mod=*/(short)0, acc8,
                           /*reuse_a=*/false, /*reuse_b=*/false);
            }
            // Scatter-add D: element v of lane is contrib[m=v+8*hi][nt*16+r]
#pragma unroll
            for (int v = 0; v < 8; ++v) {
                const int m    = v + 8 * hi;
                const int orow = __shfl(oidx, m, 32);
                if (m0 + m < M)
                    atomicAdd(&acc[(size_t)orow * 64u + (size_t)(nt * 16 + r)],
                              acc8[v]);
            }
        }
    }
}

// ---------------------------------------------------------------------------
// Kernel 2: per-channel sum & sum-of-squares over acc[n_out, 64]
// ---------------------------------------------------------------------------
__global__ void bn_stats_kernel(const float* __restrict__ acc,
                                float* __restrict__ stats, // [0:64)=sum [64:128)=sumsq
                                int n_out)
{
    const int c = threadIdx.x & 63;
    const int rowsPerBlock = blockDim.x >> 6;
    int row = blockIdx.x * rowsPerBlock + (threadIdx.x >> 6);
    const int stride = gridDim.x * rowsPerBlock;
    float s = 0.f, ss = 0.f;
    for (; row < n_out; row += stride) {
        const float v = acc[(size_t)row * 64u + c];
        s  += v;
        ss += v * v;
    }
    atomicAdd(&stats[c], s);
    atomicAdd(&stats[64 + c], ss);
}

// ---------------------------------------------------------------------------
// Kernel 3: batchnorm normalize -> d_out
// ---------------------------------------------------------------------------
__global__ void bn_apply_kernel(const float* __restrict__ acc,
                                const float* __restrict__ stats,
                                const float* __restrict__ gamma,
                                const float* __restrict__ beta,
                                float* __restrict__ y,
                                int n_out)
{
    const size_t total = (size_t)n_out * 64u;
    size_t i = (size_t)blockIdx.x * blockDim.x + threadIdx.x;
    if (i >= total) return;
    const int c = (int)(i & 63u);
    const float inv_n = 1.0f / (float)n_out;
    const float mean  = stats[c] * inv_n;
    const float var   = stats[64 + c] * inv_n - mean * mean;  // biased variance
    y[i] = (acc[i] - mean) * rsqrtf(var + 1e-5f) * gamma[c] + beta[c];
}

// ---------------------------------------------------------------------------
// Host launch
// ---------------------------------------------------------------------------
extern "C" void kernel_launch(void* const* d_in, const int* in_sizes, int n_in,
                              void* d_out, int out_size, void* d_ws, size_t ws_size,
                              hipStream_t stream)
{
    const float* x       = (const float*)d_in[0];  // [N_in, 64]
    const float* W       = (const float*)d_in[1];  // [K, 64, 64]
    const float* gamma   = (const float*)d_in[2];  // [64]
    const float* beta    = (const float*)d_in[3];  // [64]
    const int*   in_idx  = (const int*)d_in[4];    // [K, M]
    const int*   out_idx = (const int*)d_in[5];    // [K, M]
    // d_in[6] = n_out scalar on device; derive host-side from out_size instead.

    const int K     = in_sizes[1] / (64 * 64);
    const int KM    = in_sizes[4];
    const int M     = KM / K;
    const int n_out = out_size / 64;

    float* acc   = (float*)d_ws;                   // [n_out, 64] accumulator
    float* stats = acc + (size_t)n_out * 64u;      // [128] sum / sumsq

    // 0) zero accumulator + stats (deterministic per call)
    const size_t zeroFloats = (size_t)n_out * 64u + 128u;
    const size_t n4 = (zeroFloats + 3u) / 4u;
    {
        const int blk = 256;
        const int grd = (int)((n4 + blk - 1) / blk);
        zero4_kernel<<<grd, blk, 0, stream>>>((float4*)d_ws, n4);
    }

    // 1) sparse conv: TDM-stage W[k] -> gather -> WMMA -> scatter-add
    {
        const int tilesPerK = (M + 15) >> 4;
        dim3 grd((tilesPerK + TILES_PER_BLOCK - 1) / TILES_PER_BLOCK, K);
        sparse_conv_wmma_kernel<<<grd, 256, 0, stream>>>(
            x, W, in_idx, out_idx, acc, M, tilesPerK);
    }

    // 2) BN statistics
    bn_stats_kernel<<<512, 256, 0, stream>>>(acc, stats, n_out);

    // 3) BN apply
    {
        const size_t total = (size_t)n_out * 64u;
        const int blk = 256;
        const int grd = (int)((total + blk - 1) / blk);
        bn_apply_kernel<<<grd, blk, 0, stream>>>(acc, stats, gamma, beta,
                                                 (float*)d_out, n_out);
    }
}